// TopK_1245540516211
// MI455X (gfx1250) — compile-verified
//
#include <hip/hip_runtime.h>
#include <math.h>

// ---------------------------------------------------------------------------
// Pipeline: scores (WMMA f32 GEMV) -> stable radix sort (desc) -> pad fixup
//           -> gather + tanh scale + transpose.
// All buffers for the sort live in d_ws (L2-resident: ~1 MB total).
// ---------------------------------------------------------------------------

typedef __attribute__((ext_vector_type(2))) float v2f;
typedef __attribute__((ext_vector_type(8))) float v8f;

#define FDIM 512

__device__ __forceinline__ unsigned ord_of(float f) {
    unsigned x = __float_as_uint(f);
    return x ^ ((x & 0x80000000u) ? 0xFFFFFFFFu : 0x80000000u);  // ascending with float
}

// key = ~ord(score): ascending key == descending score; stable sort -> tie by index asc.
#define KEY_NEGINF 0xFF800000u  // key of -inf; score > -inf  <=>  key < KEY_NEGINF

// ---------------------------------------------------------------------------
// Kernel 1: scores[i] = dot(E[i,:], s) / ||s|| + mask[i], via V_WMMA_F32_16X16X4_F32.
// One wave computes 16 rows; B = scorer broadcast into all 16 columns, so every
// column of the 16x16 accumulator holds the same 16 row-scores.
// ---------------------------------------------------------------------------
__global__ void __launch_bounds__(256)
score_kernel(const float* __restrict__ emb, const float* __restrict__ mask,
             const float* __restrict__ scorer,
             float* __restrict__ scores, unsigned* __restrict__ keys,
             unsigned* __restrict__ pays, int nrows) {
    __shared__ float sc[FDIM];
    __shared__ float red[256];
    __shared__ float s_inv;

    const int t = threadIdx.x;
    float a0 = scorer[t];
    float a1 = scorer[t + 256];
    sc[t]       = a0;
    sc[t + 256] = a1;
    red[t] = a0 * a0 + a1 * a1;
    __syncthreads();
    for (int s = 128; s > 0; s >>= 1) {
        if (t < s) red[t] += red[t + s];
        __syncthreads();
    }
    if (t == 0) s_inv = 1.0f / sqrtf(red[0]);
    __syncthreads();

    const int wave = t >> 5;
    const int lane = t & 31;
    const int tile = blockIdx.x * 8 + wave;   // wave-uniform
    if (tile * 16 < nrows) {
        const int row0 = tile * 16;
        const int h    = lane >> 4;           // half-wave: 0 or 1
        const float* arow = emb + (size_t)(row0 + (lane & 15)) * FDIM + 2 * h;

        v8f c = {0.f, 0.f, 0.f, 0.f, 0.f, 0.f, 0.f, 0.f};
        #pragma unroll 4
        for (int k = 0; k < FDIM; k += 4) {
            // A 16x4 f32: VGPR0 = (M=lane&15, K=k+2h), VGPR1 = K=k+2h+1
            v2f a = *(const v2f*)(arow + k);
            // B 4x16 f32: VGPR0 = (K=k+h, all N), VGPR1 = (K=k+2+h, all N)
            v2f b;
            b.x = sc[k + h];
            b.y = sc[k + 2 + h];
            c = __builtin_amdgcn_wmma_f32_16x16x4_f32(
                    false, a, false, b, (short)0, c, false, false);
        }

        // Lane 0 holds D[0..7][0] (rows row0..row0+7); lane 16 holds D[8..15][0].
        if ((lane & 15) == 0) {
            const int   rb  = row0 + h * 8;
            const float inv = s_inv;
            #pragma unroll
            for (int j = 0; j < 8; ++j) {
                float s = c[j] * inv + mask[rb + j];
                scores[rb + j] = s;
                keys[rb + j]   = ~ord_of(s);
                pays[rb + j]   = (unsigned)(rb + j);
            }
        }
    }
}

// ---------------------------------------------------------------------------
// Stable LSD radix sort pass (8-bit digit). Grid = 256 blocks, one per digit.
// Phase A: full histogram (deterministic LDS atomics) -> base offset of digit.
// Phase B: stable compaction of matching elements via block prefix scans.
// ---------------------------------------------------------------------------
__device__ __forceinline__ unsigned block_incl_scan(unsigned* buf, int t, unsigned v) {
    buf[t] = v;
    __syncthreads();
    for (int off = 1; off < 256; off <<= 1) {
        unsigned x = (t >= off) ? buf[t - off] : 0u;
        __syncthreads();
        buf[t] += x;
        __syncthreads();
    }
    return buf[t];
}

__global__ void __launch_bounds__(256)
radix_pass(const unsigned* __restrict__ keyIn, const unsigned* __restrict__ payIn,
           unsigned* __restrict__ keyOut, unsigned* __restrict__ payOut,
           int n, int shift) {
    __shared__ unsigned hist[256];
    __shared__ unsigned s_base;

    const int t  = threadIdx.x;
    const int bd = blockIdx.x;  // digit owned by this block

    hist[t] = 0u;
    __syncthreads();
    for (int i = t; i < n; i += 256) {
        unsigned d = (keyIn[i] >> shift) & 255u;
        atomicAdd(&hist[d], 1u);
    }
    __syncthreads();

    // exclusive base for our digit
    (void)block_incl_scan(hist, t, hist[t] /*in-place via reload*/);
    __syncthreads();
    if (t == 0) s_base = (bd == 0) ? 0u : hist[bd - 1];
    __syncthreads();

    unsigned gbase = s_base;
    for (int c0 = 0; c0 < n; c0 += 1024) {
        const int i0 = c0 + t * 4;
        unsigned ks[4], ps[4];
        unsigned fl = 0u, cnt = 0u;
        #pragma unroll
        for (int j = 0; j < 4; ++j) {
            int i = i0 + j;
            if (i < n) {
                ks[j] = keyIn[i];
                ps[j] = payIn[i];
                if (((ks[j] >> shift) & 255u) == (unsigned)bd) {
                    fl |= 1u << j;
                    ++cnt;
                }
            }
        }
        unsigned incl  = block_incl_scan(hist, t, cnt);
        unsigned total = hist[255];
        unsigned pos   = gbase + incl - cnt;
        #pragma unroll
        for (int j = 0; j < 4; ++j) {
            if (fl & (1u << j)) {
                keyOut[pos] = ks[j];
                payOut[pos] = ps[j];
                ++pos;
            }
        }
        gbase += total;
        __syncthreads();  // hist reused next chunk
    }
}

// ---------------------------------------------------------------------------
// Pad fixup: n_valid = #{j<k : score_j > -inf} (sorted => prefix, binary search);
// last_valid = idx[max(n_valid-1,0)].
// ---------------------------------------------------------------------------
__global__ void meta_kernel(const unsigned* __restrict__ keys,
                            const unsigned* __restrict__ pays,
                            int k, int* __restrict__ meta) {
    if (threadIdx.x == 0 && blockIdx.x == 0) {
        int lo = 0, hi = k;
        while (lo < hi) {
            int mid = (lo + hi) >> 1;
            if (keys[mid] < KEY_NEGINF) lo = mid + 1; else hi = mid;
        }
        int nv = lo;
        meta[0] = nv;
        meta[1] = (int)pays[(nv > 0) ? (nv - 1) : 0];
    }
}

// ---------------------------------------------------------------------------
// Output: out[f*k + j] = E[idx_j, f] * tanh(score[idx_j]), transposed via LDS.
// 32-j tiles, two 256-f halves; padded LDS (stride 257) => conflict-free
// transposed reads; writes are 128B-contiguous per f.
// ---------------------------------------------------------------------------
__global__ void __launch_bounds__(256)
out_kernel(const float* __restrict__ emb, const float* __restrict__ scores,
           const unsigned* __restrict__ sidx, const int* __restrict__ meta,
           float* __restrict__ out, int k) {
    __shared__ float tile[32][257];
    __shared__ int   ridx[32];
    __shared__ float scal[32];

    const int t  = threadIdx.x;
    const int j0 = blockIdx.x * 32;

    if (t < 32) {
        int j = j0 + t;
        if (j < k) {
            int nv = meta[0];
            int id = (j < nv) ? (int)sidx[j] : meta[1];
            ridx[t] = id;
            scal[t] = tanhf(scores[id]);
        } else {
            ridx[t] = -1;
        }
    }
    __syncthreads();

    for (int half = 0; half < 2; ++half) {
        const int fbase = half * 256;
        // load: 32 rows x 256 f = 2048 float4 gathers (coalesced within a row)
        for (int e = t; e < 2048; e += 256) {
            int r = e >> 6;   // row in tile
            int q = e & 63;   // float4 index within 256-f half
            if (ridx[r] >= 0) {
                float4 v = *(const float4*)(emb + (size_t)ridx[r] * FDIM + fbase + q * 4);
                float  s = scal[r];
                tile[r][q * 4 + 0] = v.x * s;
                tile[r][q * 4 + 1] = v.y * s;
                tile[r][q * 4 + 2] = v.z * s;
                tile[r][q * 4 + 3] = v.w * s;
            }
        }
        __syncthreads();
        // store transposed: each wave owns f = w, w+8, ... ; 32 contiguous j per f
        const int jj = t & 31;
        const int j  = j0 + jj;
        for (int f = (t >> 5); f < 256; f += 8) {
            if (j < k) out[(size_t)(fbase + f) * k + j] = tile[jj][f];
        }
        __syncthreads();
    }
}

// ---------------------------------------------------------------------------
extern "C" void kernel_launch(void* const* d_in, const int* in_sizes, int n_in,
                              void* d_out, int out_size, void* d_ws, size_t ws_size,
                              hipStream_t stream) {
    const float* emb    = (const float*)d_in[0];   // [N, F]
    const float* mask   = (const float*)d_in[1];   // [N, 1]
    const float* scorer = (const float*)d_in[2];   // [F, 1]
    const int N = in_sizes[1];
    const int F = in_sizes[2];                     // 512
    const int k = out_size / F;                    // 25000
    float* out = (float*)d_out;

    // workspace layout (each slot 16B-aligned)
    const int slot = ((N + 255) / 256) * 256;
    char* ws = (char*)d_ws;
    float*    scores = (float*)ws;
    unsigned* keyA   = (unsigned*)(ws + (size_t)slot * 4);
    unsigned* keyB   = keyA + slot;
    unsigned* payA   = keyB + slot;
    unsigned* payB   = payA + slot;
    int*      meta   = (int*)(payB + slot);

    const int tiles   = N / 16;                    // N = 50000 -> 3125
    const int blocksA = (tiles + 7) / 8;
    score_kernel<<<blocksA, 256, 0, stream>>>(emb, mask, scorer, scores, keyA, payA, N);

    radix_pass<<<256, 256, 0, stream>>>(keyA, payA, keyB, payB, N, 0);
    radix_pass<<<256, 256, 0, stream>>>(keyB, payB, keyA, payA, N, 8);
    radix_pass<<<256, 256, 0, stream>>>(keyA, payA, keyB, payB, N, 16);
    radix_pass<<<256, 256, 0, stream>>>(keyB, payB, keyA, payA, N, 24);

    meta_kernel<<<1, 1, 0, stream>>>(keyA, payA, k, meta);

    const int blocksO = (k + 31) / 32;
    out_kernel<<<blocksO, 256, 0, stream>>>(emb, scores, payA, meta, out, k);
}